// XSReLU_cw_perc_param_2_47528108097998
// MI455X (gfx1250) — compile-verified
//
#include <hip/hip_runtime.h>
#include <hip/hip_bf16.h>

// XSReLU channel-wise percentile op.
// Shapes: input (64, 256, 64, 64) f32  -> rows = 16384, n = 4096 per row.
// Per row: need the i_low-th and i_high-th smallest elements (order stats),
// then out = relu(x - x_low) + (relu(x - x_high) - relu(x - x_low)) * p.
//
// Strategy: 1 workgroup (256 threads, 8 wave32) per row.
//   - async-copy the 16 KB row global -> LDS (CDNA5 GLOBAL_LOAD_ASYNC_TO_LDS_B128)
//   - radix-select (8-bit digits, 4 passes) twice in LDS for the two ranks;
//     crossing-bin search done by a single wave32 with __shfl_up scan
//     (no block barriers inside the scan) -> 3 barriers per pass total
//   - elementwise blend from LDS, nontemporal b128 stores to global
// Single global read + single global write = 512 MB total -> ~22 us at 23.3 TB/s.

#define N_PER_ROW 4096
#define BLOCK 256
#define VEC_ITERS ((N_PER_ROW / 4) / BLOCK)   // 4

#if defined(__has_builtin)
#  if __has_builtin(__builtin_amdgcn_global_load_async_to_lds_b128)
#    define HAVE_ASYNC_LDS 1
#  endif
#  if __has_builtin(__builtin_amdgcn_s_wait_asynccnt)
#    define HAVE_WAIT_ASYNC 1
#  endif
#endif

// Native clang ext-vectors (HIP_vector_type is rejected by some builtins).
typedef float v4f __attribute__((ext_vector_type(4)));
typedef int   v4i __attribute__((ext_vector_type(4)));
// Async builtin takes 'int4 __device__*' (AS1) and 'int4 __shared__*' (AS3).
typedef __attribute__((address_space(1))) v4i g_v4i;
typedef __attribute__((address_space(3))) v4i l_v4i;

// Monotone bijection float -> uint32 (preserves total order, -0 < +0).
__device__ __forceinline__ unsigned f2key(float f) {
  unsigned u = __float_as_uint(f);
  return u ^ ((u & 0x80000000u) ? 0xFFFFFFFFu : 0x80000000u);
}
__device__ __forceinline__ float key2f(unsigned k) {
  unsigned u = (k & 0x80000000u) ? (k ^ 0x80000000u) : ~k;
  return __uint_as_float(u);
}

// Cooperative (whole block) radix select of the k-th smallest element of the
// n=4096 floats resident in s_data. Returns identical value in all threads.
// The 256-bin crossing search is done entirely inside wave 0 (lanes 0..31,
// 8 bins/lane, wave32 shfl_up inclusive scan) -- no barriers inside the scan.
__device__ float radix_select(const float* s_data, unsigned* s_hist,
                              unsigned* s_found, int k, int tid) {
  unsigned prefix = 0u, himask = 0u;
  for (int shift = 24; shift >= 0; shift -= 8) {
    s_hist[tid] = 0u;                      // 256 threads <-> 256 bins
    __syncthreads();                       // barrier 1: hist cleared
    #pragma unroll
    for (int j = 0; j < N_PER_ROW / BLOCK; ++j) {
      unsigned key = f2key(s_data[tid + j * BLOCK]);
      if ((key & himask) == prefix)
        atomicAdd(&s_hist[(key >> shift) & 0xFFu], 1u);
    }
    __syncthreads();                       // barrier 2: hist complete

    if (tid < 32) {                        // wave 0 finds the crossing bin
      const unsigned base = (unsigned)tid << 3;
      unsigned h[8];
      unsigned gsum = 0u;
      #pragma unroll
      for (int j = 0; j < 8; ++j) { h[j] = s_hist[base + j]; gsum += h[j]; }
      // wave32 inclusive scan of the 32 group sums (no LDS, no barriers)
      unsigned incl = gsum;
      #pragma unroll
      for (int off = 1; off < 32; off <<= 1) {
        unsigned t = __shfl_up(incl, off, 32);
        if (tid >= off) incl += t;
      }
      const unsigned excl = incl - gsum;
      if ((unsigned)k >= excl && (unsigned)k < incl) {  // exactly one lane
        unsigned c = excl;
        #pragma unroll
        for (int j = 0; j < 8; ++j) {
          if ((unsigned)k < c + h[j]) {
            s_found[0] = base + (unsigned)j;   // winning digit
            s_found[1] = (unsigned)k - c;      // rank within digit
            break;
          }
          c += h[j];
        }
      }
    }
    __syncthreads();                       // barrier 3: s_found published
    prefix |= s_found[0] << shift;
    himask |= 0xFFu << shift;
    k = (int)s_found[1];
  }
  return key2f(prefix);
}

__global__ __launch_bounds__(BLOCK) void xsrelu_percentile_kernel(
    const float* __restrict__ in, const float* __restrict__ plogit,
    float* __restrict__ out) {
  __shared__ __align__(16) float s_data[N_PER_ROW];   // 16 KB row stage
  __shared__ unsigned s_hist[256];
  __shared__ unsigned s_found[2];

  const int tid = threadIdx.x;
  const long long row = blockIdx.x;
  const float* src = in + row * (long long)N_PER_ROW;
  float* dst = out + row * (long long)N_PER_ROW;

  // ---- Stage the row into LDS (async global->LDS path on CDNA5) ----
#if defined(HAVE_ASYNC_LDS)
  #pragma unroll
  for (int j = 0; j < VEC_ITERS; ++j) {
    const int i = tid + j * BLOCK;        // float4 index, constant trip count
    __builtin_amdgcn_global_load_async_to_lds_b128(
        (g_v4i*)(src + 4 * i), (l_v4i*)(&s_data[4 * i]),
        /*offset=*/0, /*cpol=*/0);
  }
#  if defined(HAVE_WAIT_ASYNC)
  __builtin_amdgcn_s_wait_asynccnt(0);
#  else
  asm volatile("s_wait_asynccnt 0" ::: "memory");
#  endif
#else
  #pragma unroll
  for (int j = 0; j < VEC_ITERS; ++j) {
    const int i = tid + j * BLOCK;
    ((v4f*)s_data)[i] = ((const v4f*)src)[i];
  }
#endif
  __syncthreads();

  // ---- Percentile parameters (uniform across block) ----
  const float p = 1.0f / (1.0f + expf(-plogit[0]));
  int i_low  = (int)floorf((float)N_PER_ROW * (p - 0.01f));
  int i_high = (int)floorf((float)N_PER_ROW * (p + 0.01f));
  i_low  = min(max(i_low,  0), N_PER_ROW - 1);
  i_high = min(max(i_high, 0), N_PER_ROW - 1);

  // ---- Two order statistics via LDS radix select ----
  const float x_low  = radix_select(s_data, s_hist, s_found, i_low,  tid);
  __syncthreads();
  const float x_high = radix_select(s_data, s_hist, s_found, i_high, tid);

  // ---- Elementwise blend, streamed out with nontemporal b128 stores ----
  const v4f* sv = (const v4f*)s_data;
  v4f* dv = (v4f*)dst;
  #pragma unroll
  for (int j = 0; j < VEC_ITERS; ++j) {
    const int i = tid + j * BLOCK;
    v4f x = sv[i];
    v4f r;
    {
      float rl = fmaxf(x.x - x_low, 0.0f), rh = fmaxf(x.x - x_high, 0.0f);
      r.x = rl + (rh - rl) * p;
    }
    {
      float rl = fmaxf(x.y - x_low, 0.0f), rh = fmaxf(x.y - x_high, 0.0f);
      r.y = rl + (rh - rl) * p;
    }
    {
      float rl = fmaxf(x.z - x_low, 0.0f), rh = fmaxf(x.z - x_high, 0.0f);
      r.z = rl + (rh - rl) * p;
    }
    {
      float rl = fmaxf(x.w - x_low, 0.0f), rh = fmaxf(x.w - x_high, 0.0f);
      r.w = rl + (rh - rl) * p;
    }
    __builtin_nontemporal_store(r, &dv[i]);
  }
}

extern "C" void kernel_launch(void* const* d_in, const int* in_sizes, int n_in,
                              void* d_out, int out_size, void* d_ws, size_t ws_size,
                              hipStream_t stream) {
  const float* in     = (const float*)d_in[0];   // (64,256,64,64) f32
  const float* plogit = (const float*)d_in[1];   // (1,) f32
  float* out = (float*)d_out;

  const int rows = out_size / N_PER_ROW;         // 16384
  xsrelu_percentile_kernel<<<dim3(rows), dim3(BLOCK), 0, stream>>>(in, plogit, out);
}